// TRANSSTNBLOCK_44083544326620
// MI455X (gfx1250) — compile-verified
//
#include <hip/hip_runtime.h>
#include <hip/hip_bf16.h>

// TransSTN block (2x deformable conv) for MI455X / gfx1250.
// Heavy math = two implicit GEMMs (256 x 2304 x 16384 f32) executed with
// V_WMMA_F32_16X16X4_F32 (exact f32 matrix pipe). Bilinear sampling params are
// precomputed once per (k,pixel) since they are channel-invariant.
// Workspace requirement: ~21.5 MB.

#define HH 64
#define WW 64
#define HWP 4096            // 64*64
#define NB  4
#define CO  256
#define PTOT 16384          // NB*HWP

typedef float v2f __attribute__((ext_vector_type(2)));
typedef float v8f __attribute__((ext_vector_type(8)));

// ---------------------------------------------------------------------------
// Kernel 1: 6-channel offset conv (tm:4, tr:2) + affine offset -> sample table
// samp[k*PTOT + pix] = { floor(py), floor(px), wy, wx }
// ---------------------------------------------------------------------------
__global__ __launch_bounds__(256) void offset_kernel(
    const float* __restrict__ xin,
    const float* __restrict__ tmw, const float* __restrict__ tmb,
    const float* __restrict__ trw, const float* __restrict__ trb,
    float4* __restrict__ samp, int Cin)
{
    const int pixg = blockIdx.x * 256 + threadIdx.x;     // 0..16383
    const int n  = pixg >> 12;
    const int hw = pixg & 4095;
    const int h  = hw >> 6;
    const int w  = hw & 63;

    float acc[6];
    acc[0] = tmb[0]; acc[1] = tmb[1]; acc[2] = tmb[2]; acc[3] = tmb[3];
    acc[4] = trb[0]; acc[5] = trb[1];

    for (int c = 0; c < Cin; ++c) {
        const float* xp = xin + ((size_t)n * Cin + c) * HWP;
        float nv[9];
#pragma unroll
        for (int k = 0; k < 9; ++k) {
            const int y = h + k / 3 - 1;
            const int x = w + k % 3 - 1;
            nv[k] = (y >= 0 && y < HH && x >= 0 && x < WW) ? xp[y * WW + x] : 0.0f;
        }
#pragma unroll
        for (int o = 0; o < 4; ++o) {
            const float* wt = tmw + ((size_t)o * Cin + c) * 9;
            float s = 0.0f;
#pragma unroll
            for (int k = 0; k < 9; ++k) s += nv[k] * wt[k];
            acc[o] += s;
        }
#pragma unroll
        for (int o = 0; o < 2; ++o) {
            const float* wt = trw + ((size_t)o * Cin + c) * 9;
            float s = 0.0f;
#pragma unroll
            for (int k = 0; k < 9; ++k) s += nv[k] * wt[k];
            acc[4 + o] += s;
        }
    }

    // off_y = tm0*dy + tm1*dx - dy + tr0 ; off_x = tm2*dy + tm3*dx - dx + tr1
#pragma unroll
    for (int k = 0; k < 9; ++k) {
        const float dy = (float)(k / 3 - 1);
        const float dx = (float)(k % 3 - 1);
        const float offy = acc[0] * dy + acc[1] * dx - dy + acc[4];
        const float offx = acc[2] * dy + acc[3] * dx - dx + acc[5];
        const float py = (float)h + dy + offy;
        const float px = (float)w + dx + offx;
        const float y0 = floorf(py);
        const float x0 = floorf(px);
        samp[k * PTOT + pixg] = make_float4(y0, x0, py - y0, px - x0);
    }
}

// ---------------------------------------------------------------------------
// Kernel 2: weight reorder (Co,C,3,3) -> (k, Co, C) for coalesced A-tile loads
// ---------------------------------------------------------------------------
__global__ __launch_bounds__(256) void reorder_w_kernel(
    const float* __restrict__ wsrc, float* __restrict__ wdst, int Cin)
{
    const int idx = blockIdx.x * 256 + threadIdx.x;      // 0 .. Co*Cin-1
    const int co = idx / Cin;
    const int c  = idx - co * Cin;
#pragma unroll
    for (int k = 0; k < 9; ++k)
        wdst[((size_t)k * CO + co) * Cin + c] = wsrc[((size_t)co * Cin + c) * 9 + k];
}

// ---------------------------------------------------------------------------
// Kernel 3: deformable conv as implicit GEMM with V_WMMA_F32_16X16X4_F32.
// Block = 256 threads = 8 waves, arranged 4 (Co) x 2 (pixel).
// Output tile per block: 64 Co x 32 pixels. Reduction K = Cin*9.
// ---------------------------------------------------------------------------
__global__ __launch_bounds__(256) void deform_wmma_gemm(
    const float* __restrict__ xin,          // (N, Cin, H, W)
    const float4* __restrict__ samp,        // (9, PTOT)
    const float* __restrict__ wr,           // (9, CO, Cin)
    float* __restrict__ outp,               // (N, CO, H, W) raw conv output
    int Cin)
{
    __shared__ float As[64][33];            // A tile: 64 Co x 32 channel-chunk
    __shared__ float Bs[32][33];            // B tile: 32 channel-chunk x 32 pix

    const int tid  = threadIdx.x;
    const int wave = tid >> 5;
    const int lane = tid & 31;
    const int half = lane >> 4;
    const int lm   = lane & 15;
    const int co_w  = wave & 3;             // 4 Co sub-tiles of 16
    const int pix_w = wave >> 2;            // 2 pixel sub-tiles of 16

    const int pixBase = blockIdx.x * 32;    // 32 pixels, all in same image n
    const int coBase  = blockIdx.y * 64;
    const int n       = pixBase >> 12;
    const float* xn   = xin + (size_t)n * Cin * HWP;

    v8f acc0 = {0.f, 0.f, 0.f, 0.f, 0.f, 0.f, 0.f, 0.f};
    v8f acc1 = {0.f, 0.f, 0.f, 0.f, 0.f, 0.f, 0.f, 0.f};

    for (int k = 0; k < 9; ++k) {
        // Per-thread sampling state for its fixed pixel (p = lane), fixed per k.
        const float4 s = samp[k * PTOT + pixBase + lane];
        const int y0 = (int)s.x, x0 = (int)s.y;
        const float wy = s.z, wx = s.w;
        const int y0c = min(max(y0, 0), HH - 1), y1c = min(max(y0 + 1, 0), HH - 1);
        const int x0c = min(max(x0, 0), WW - 1), x1c = min(max(x0 + 1, 0), WW - 1);
        const float vy0 = (y0 >= 0 && y0 < HH) ? 1.f : 0.f;
        const float vy1 = (y0 + 1 >= 0 && y0 + 1 < HH) ? 1.f : 0.f;
        const float vx0 = (x0 >= 0 && x0 < WW) ? 1.f : 0.f;
        const float vx1 = (x0 + 1 >= 0 && x0 + 1 < WW) ? 1.f : 0.f;
        const float w00 = (1.f - wy) * (1.f - wx) * vy0 * vx0;
        const float w01 = (1.f - wy) * wx         * vy0 * vx1;
        const float w10 = wy         * (1.f - wx) * vy1 * vx0;
        const float w11 = wy         * wx         * vy1 * vx1;
        const int i00 = y0c * WW + x0c, i01 = y0c * WW + x1c;
        const int i10 = y1c * WW + x0c, i11 = y1c * WW + x1c;

        const float* wk = wr + (size_t)k * CO * Cin;

        for (int c0 = 0; c0 < Cin; c0 += 32) {
            __syncthreads();
            // Stage A tile: 64x32 weights (lane-contiguous in channel).
#pragma unroll
            for (int j = 0; j < 8; ++j) {
                const int co = wave + 8 * j;
                As[co][lane] = wk[(size_t)(coBase + co) * Cin + c0 + lane];
            }
            // Stage B tile: bilinear-gathered im2col column chunk.
#pragma unroll
            for (int j = 0; j < 4; ++j) {
                const int cl = wave + 8 * j;
                const float* xp = xn + (size_t)(c0 + cl) * HWP;
                Bs[cl][lane] = w00 * xp[i00] + w01 * xp[i01]
                             + w10 * xp[i10] + w11 * xp[i11];
            }
            __syncthreads();

            // 8 x K=4 WMMA steps; two accumulators break the D->C RAW chain.
#pragma unroll
            for (int st = 0; st < 8; ++st) {
                const int kk = st * 4 + 2 * half;   // f32 A/B fragment layout
                v2f a, b;
                a.x = As[co_w * 16 + lm][kk];
                a.y = As[co_w * 16 + lm][kk + 1];
                b.x = Bs[kk][pix_w * 16 + lm];
                b.y = Bs[kk + 1][pix_w * 16 + lm];
                if (st & 1)
                    acc1 = __builtin_amdgcn_wmma_f32_16x16x4_f32(
                        false, a, false, b, (short)0, acc1, false, false);
                else
                    acc0 = __builtin_amdgcn_wmma_f32_16x16x4_f32(
                        false, a, false, b, (short)0, acc0, false, false);
            }
        }
    }

    const v8f acc = acc0 + acc1;
    const int pixg = pixBase + pix_w * 16 + lm;
    const int hw   = pixg & 4095;
#pragma unroll
    for (int r = 0; r < 8; ++r) {
        const int co = coBase + co_w * 16 + r + 8 * half;   // C/D VGPR layout
        outp[((size_t)n * CO + co) * HWP + hw] = acc[r];
    }
}

// ---------------------------------------------------------------------------
// Kernel 4: per-channel BN stats -> fused scale/shift
// ---------------------------------------------------------------------------
__global__ __launch_bounds__(256) void bn_stats_kernel(
    const float* __restrict__ raw,
    const float* __restrict__ g, const float* __restrict__ b,
    float* __restrict__ scale, float* __restrict__ shift)
{
    const int c = blockIdx.x;
    const int tid = threadIdx.x;
    float s = 0.f, sq = 0.f;
    for (int n = 0; n < NB; ++n) {
        const float* p = raw + ((size_t)n * CO + c) * HWP;
        for (int i = tid; i < HWP; i += 256) {
            const float v = p[i];
            s += v; sq += v * v;
        }
    }
    __shared__ float rs[256];
    __shared__ float rq[256];
    rs[tid] = s; rq[tid] = sq;
    __syncthreads();
    for (int off = 128; off > 0; off >>= 1) {
        if (tid < off) { rs[tid] += rs[tid + off]; rq[tid] += rq[tid + off]; }
        __syncthreads();
    }
    if (tid == 0) {
        const float mean = rs[0] * (1.f / 16384.f);
        const float var  = rq[0] * (1.f / 16384.f) - mean * mean;
        const float sc   = g[c] * rsqrtf(var + 1e-5f);
        scale[c] = sc;
        shift[c] = b[c] - mean * sc;
    }
}

// ---------------------------------------------------------------------------
// Kernel 5: BN apply + relu (+ optional residual)
// ---------------------------------------------------------------------------
__global__ __launch_bounds__(256) void bn_apply_kernel(
    const float* __restrict__ raw,
    const float* __restrict__ scale, const float* __restrict__ shift,
    const float* __restrict__ residual,
    float* __restrict__ outp)
{
    const size_t idx = (size_t)blockIdx.x * 256 + threadIdx.x;  // < 4194304
    const int c = ((int)(idx >> 12)) & 255;
    float v = raw[idx] * scale[c] + shift[c];
    if (residual) v += residual[idx];
    outp[idx] = fmaxf(v, 0.f);
}

// ---------------------------------------------------------------------------
extern "C" void kernel_launch(void* const* d_in, const int* in_sizes, int n_in,
                              void* d_out, int out_size, void* d_ws, size_t ws_size,
                              hipStream_t stream) {
    const float* x     = (const float*)d_in[0];
    const float* tm1w  = (const float*)d_in[1];
    const float* tm1b  = (const float*)d_in[2];
    const float* tr1w  = (const float*)d_in[3];
    const float* tr1b  = (const float*)d_in[4];
    const float* dcn1w = (const float*)d_in[5];
    const float* bn1g  = (const float*)d_in[6];
    const float* bn1b  = (const float*)d_in[7];
    const float* tm2w  = (const float*)d_in[8];
    const float* tm2b  = (const float*)d_in[9];
    const float* tr2w  = (const float*)d_in[10];
    const float* tr2b  = (const float*)d_in[11];
    const float* dcn2w = (const float*)d_in[12];
    const float* bn2g  = (const float*)d_in[13];
    const float* bn2b  = (const float*)d_in[14];
    float* out = (float*)d_out;

    char* ws = (char*)d_ws;
    float*  raw   = (float*)(ws);                       // 16 MB raw conv out
    float4* samp  = (float4*)(ws + 16777216);           // 2.25 MB sample table
    float*  wr    = (float*)(ws + 19136512);            // 2.25 MB reordered W
    float*  scale = (float*)(ws + 21495808);            // 1 KB
    float*  shift = (float*)(ws + 21496832);            // 1 KB

    // ---- Stage 1: input = x, activation written into d_out ----
    offset_kernel<<<64, 256, 0, stream>>>(x, tm1w, tm1b, tr1w, tr1b, samp, 256);
    reorder_w_kernel<<<256, 256, 0, stream>>>(dcn1w, wr, 256);
    deform_wmma_gemm<<<dim3(512, 4), 256, 0, stream>>>(x, samp, wr, raw, 256);
    bn_stats_kernel<<<256, 256, 0, stream>>>(raw, bn1g, bn1b, scale, shift);
    bn_apply_kernel<<<16384, 256, 0, stream>>>(raw, scale, shift, nullptr, out);

    // ---- Stage 2: input = d_out (act1), residual = x, final -> d_out ----
    offset_kernel<<<64, 256, 0, stream>>>(out, tm2w, tm2b, tr2w, tr2b, samp, 256);
    reorder_w_kernel<<<256, 256, 0, stream>>>(dcn2w, wr, 256);
    deform_wmma_gemm<<<dim3(512, 4), 256, 0, stream>>>(out, samp, wr, raw, 256);
    bn_stats_kernel<<<256, 256, 0, stream>>>(raw, bn2g, bn2b, scale, shift);
    bn_apply_kernel<<<16384, 256, 0, stream>>>(raw, scale, shift, x, out);
}